// VectorQuantizer_17600775979270
// MI455X (gfx1250) — compile-verified
//
#include <hip/hip_runtime.h>

// ---- problem constants --------------------------------------------------
#define BB   8
#define DD   128
#define TT   4096
#define NC   1024
#define BT   (BB * TT)           // 32768 rows
#define MROWS 64                 // rows per block (4 x 16 M-subtiles)
#define MBLOCKS (BT / MROWS)     // 512 blocks

// flat output offsets (floats), in reference return order
#define ZQ_OFF    ((size_t)0)
#define CODES_OFF ((size_t)BB * DD * TT)            // 4194304
#define LOSS_OFF  (CODES_OFF + (size_t)BT)          // 4227072
#define DIST_OFF  (LOSS_OFF + 1)                    // 4227073

typedef __attribute__((ext_vector_type(2))) float v2f;
typedef __attribute__((ext_vector_type(8))) float v8f;

// ---- kernel 1: codebook squared norms into ws[0..1024) ------------------
__global__ __launch_bounds__(256) void vq_cnorm(const float* __restrict__ cb,
                                                float* __restrict__ cnorm) {
  int n = blockIdx.x * 256 + threadIdx.x;
  if (n < NC) {
    const float* p = cb + (size_t)n * DD;
    float s = 0.f;
#pragma unroll 4
    for (int d = 0; d < DD; ++d) s += p[d] * p[d];
    cnorm[n] = s;
  }
}

// ---- async stage of one 128-code codebook slab into an LDS buffer -------
// slab j covers codes [128j, 128j+128); 256 threads x 16 b128 each = 64KB.
// Uses CDNA5 GLOBAL_LOAD_ASYNC_TO_LDS_B128 (ASYNCcnt-tracked).
__device__ __forceinline__ void stage_slab(const float* __restrict__ cb, int j,
                                           float (*bsh)[132], int tid) {
  const float* src = cb + (size_t)(128 * j) * DD;
  unsigned ldsBase = (unsigned)(uintptr_t)(&bsh[0][0]);   // low 32 bits = LDS offset
#pragma unroll
  for (int it = 0; it < 16; ++it) {
    int q = tid + it * 256;          // float4 index within 128x128 slab
    int n = q >> 5;                  // code row within slab
    int d = (q & 31) << 2;           // dim, multiple of 4 (16B aligned; 132*4B row stride is 16B aligned)
    unsigned laddr = ldsBase + (unsigned)((n * 132 + d) * 4);
    const float* g = src + (size_t)n * DD + d;
    asm volatile("global_load_async_to_lds_b128 %0, %1, off"
                 :: "v"(laddr), "v"(g) : "memory");
  }
}

// ---- kernel 2: main VQ kernel -------------------------------------------
// block = 256 threads (8 waves), 64 flat rows per block.
// wave w handles codes n0 = 128j + 16w per slab j, for all 4 M-subtiles.
__global__ __launch_bounds__(256) void vq_main(const float* __restrict__ z,
                                               const float* __restrict__ cb,
                                               const float* __restrict__ cnorm,
                                               float* __restrict__ out,
                                               float* __restrict__ lossPartial) {
  __shared__ float As[MROWS][132];     // z tile (33.8KB), pad -> bank step 4
  __shared__ float Bsh[2][128][132];   // double-buffered codebook slab (135.2KB)
  __shared__ float Zn[MROWS];
  __shared__ float WMinV[8][MROWS];
  __shared__ int   WMinI[8][MROWS];
  __shared__ int   SCode[MROWS];
  __shared__ float Red[256];

  const int tid  = threadIdx.x;
  const int lane = tid & 31;
  const int wave = tid >> 5;
  const int hi   = lane >> 4;          // lane half (K split in A/B frags, M split in C)
  const int ln   = lane & 15;

  const int m0 = blockIdx.x * MROWS;
  const int b  = m0 >> 12;             // 64 | 4096, tiles never cross batches
  const int t0 = m0 & 4095;
  const float* zb = z + (size_t)b * DD * TT;

  // ---- stage z tile (64 rows x 128 dims), coalesced over t --------------
#pragma unroll
  for (int i = 0; i < 32; ++i) {
    int idx = tid + i * 256;           // 0..8191
    int row = idx & 63;
    int d   = idx >> 6;
    As[row][d] = zb[(size_t)d * TT + t0 + row];
  }
  __syncthreads();
  if (tid < MROWS) {
    float s = 0.f;
#pragma unroll 4
    for (int d = 0; d < DD; ++d) { float v = As[tid][d]; s += v * v; }
    Zn[tid] = s;
  }

  // prologue: async-fetch slab 0 into buffer 0
  stage_slab(cb, 0, Bsh[0], tid);
  asm volatile("s_wait_asynccnt 0x0" ::: "memory");
  __syncthreads();                     // slab 0 + Zn visible to all waves

  float minv[4][8];
  int   mini[4][8];
#pragma unroll
  for (int s = 0; s < 4; ++s)
#pragma unroll
    for (int v = 0; v < 8; ++v) { minv[s][v] = 3.4e38f; mini[s][v] = 0; }

  float* dist = out + DIST_OFF;

  for (int j = 0; j < 8; ++j) {
    const int cur = j & 1;
    // overlap: stream slab j+1 from L2 into the other buffer while computing j
    if (j + 1 < 8) stage_slab(cb, j + 1, Bsh[cur ^ 1], tid);

    const int n0 = 128 * j + 16 * wave;            // ascending in j per lane -> argmin tie-break ok
    const float cn = cnorm[n0 + ln];

    v8f acc[4] = {v8f{}, v8f{}, v8f{}, v8f{}};
    const float (* __restrict__ bt)[132] = Bsh[cur];

#pragma unroll 4
    for (int k0 = 0; k0 < DD; k0 += 4) {
      const int kb = k0 + 2 * hi;                  // ISA A/B fragment K split
      v2f bf;
      bf.x = bt[16 * wave + ln][kb];
      bf.y = bt[16 * wave + ln][kb + 1];
#pragma unroll
      for (int s = 0; s < 4; ++s) {                // B reused across 4 M-subtiles
        v2f a;
        a.x = As[16 * s + ln][kb];
        a.y = As[16 * s + ln][kb + 1];
        acc[s] = __builtin_amdgcn_wmma_f32_16x16x4_f32(
            false, a, false, bf, (short)0, acc[s], false, false);
      }
    }

    // dist = ||z||^2 - 2P + ||c||^2 ; store + running argmin
#pragma unroll
    for (int s = 0; s < 4; ++s) {
#pragma unroll
      for (int v = 0; v < 8; ++v) {
        const int r = 16 * s + v + 8 * hi;         // C layout: row = v + 8*laneHalf
        float dv = Zn[r] + cn - 2.0f * acc[s][v];
        dist[(size_t)(m0 + r) * NC + (n0 + ln)] = dv;
        if (dv < minv[s][v]) { minv[s][v] = dv; mini[s][v] = n0 + ln; }
      }
    }

    asm volatile("s_wait_asynccnt 0x0" ::: "memory");  // slab j+1 landed
    __syncthreads();                                   // all waves done with buf cur
  }

  // ---- argmin reduce across the 16 lanes of each half-wave --------------
#pragma unroll
  for (int s = 0; s < 4; ++s)
#pragma unroll
    for (int v = 0; v < 8; ++v) {
      float bv = minv[s][v];
      int   bi = mini[s][v];
#pragma unroll
      for (int mask = 8; mask >= 1; mask >>= 1) {
        float ov = __shfl_xor(bv, mask, 32);
        int   oi = __shfl_xor(bi, mask, 32);
        if (ov < bv || (ov == bv && oi < bi)) { bv = ov; bi = oi; }
      }
      if (ln == 0) {
        WMinV[wave][16 * s + v + 8 * hi] = bv;
        WMinI[wave][16 * s + v + 8 * hi] = bi;
      }
    }
  __syncthreads();

  // ---- reduce across the 8 waves, write codes ---------------------------
  if (tid < MROWS) {
    float bv = WMinV[0][tid];
    int   bi = WMinI[0][tid];
#pragma unroll
    for (int w = 1; w < 8; ++w) {
      float v = WMinV[w][tid];
      int   i2 = WMinI[w][tid];
      if (v < bv || (v == bv && i2 < bi)) { bv = v; bi = i2; }
    }
    SCode[tid] = bi;
    out[CODES_OFF + m0 + tid] = (float)bi;   // harness output buffer is f32
  }
  __syncthreads();

  // ---- gather z_q, write transposed (B,D,T), accumulate MSE -------------
  float ls = 0.f;
#pragma unroll
  for (int i = 0; i < 32; ++i) {
    int idx = tid + i * 256;
    int row = idx & 63;
    int d   = idx >> 6;
    float qv = cb[(size_t)SCode[row] * DD + d];
    float zv = As[row][d];
    float df = zv - qv;
    ls += df * df;
    // z_q_st == z_q numerically; output layout (B, D, T)
    out[ZQ_OFF + (size_t)b * DD * TT + (size_t)d * TT + (t0 + row)] = qv;
  }
  Red[tid] = ls;
  __syncthreads();
#pragma unroll
  for (int st = 128; st > 0; st >>= 1) {
    if (tid < st) Red[tid] += Red[tid + st];
    __syncthreads();
  }
  if (tid == 0) lossPartial[blockIdx.x] = Red[0];
}

// ---- kernel 3: deterministic final loss reduction -----------------------
__global__ __launch_bounds__(256) void vq_loss(const float* __restrict__ partial,
                                               float* __restrict__ out) {
  __shared__ float Red[256];
  float s = 0.f;
  for (int i = threadIdx.x; i < MBLOCKS; i += 256) s += partial[i];
  Red[threadIdx.x] = s;
  __syncthreads();
#pragma unroll
  for (int st = 128; st > 0; st >>= 1) {
    if (threadIdx.x < st) Red[threadIdx.x] += Red[threadIdx.x + st];
    __syncthreads();
  }
  // loss = codebk + 0.25*commit = 1.25 * mean((z - z_q)^2)
  if (threadIdx.x == 0) out[LOSS_OFF] = 1.25f * Red[0] / (float)((size_t)BB * TT * DD);
}

// ---- launcher -----------------------------------------------------------
extern "C" void kernel_launch(void* const* d_in, const int* in_sizes, int n_in,
                              void* d_out, int out_size, void* d_ws, size_t ws_size,
                              hipStream_t stream) {
  (void)in_sizes; (void)n_in; (void)out_size; (void)ws_size;
  const float* z  = (const float*)d_in[0];   // (8, 128, 4096) f32
  const float* cb = (const float*)d_in[1];   // (1024, 128) f32
  float* out = (float*)d_out;

  float* cnorm   = (float*)d_ws;             // 1024 floats
  float* partial = (float*)d_ws + NC;        // 512 floats

  vq_cnorm<<<(NC + 255) / 256, 256, 0, stream>>>(cb, cnorm);
  vq_main<<<MBLOCKS, 256, 0, stream>>>(z, cb, cnorm, out, partial);
  vq_loss<<<1, 256, 0, stream>>>(partial, out);
}